// TradeNet_7301444403777
// MI455X (gfx1250) — compile-verified
//
#include <hip/hip_runtime.h>
#include <stdint.h>

#define T_LEN   524288
#define NSRC    32
#define NTGT    32
#define WIN     64
#define KDIM    (NSRC * WIN)          // 2048
#define L_OUT   (T_LEN - WIN + 1)     // 524225
#define OUTW    (L_OUT + 1)           // 524226
#define CHUNKS  2048
#define CLEN    256                   // CHUNKS*CLEN = 524288 >= L_OUT
#define TILE_T  256                   // time outputs per block (16 N-tiles)
#define XCOLS   (TILE_T + WIN - 1)    // 319 staged x columns
#define XSTRIDE 324                   // padded LDS row stride (floats)

typedef __attribute__((ext_vector_type(2))) float v2f;
typedef __attribute__((ext_vector_type(8))) float v8f;

// ---------------------------------------------------------------------------
// Stage 1: implicit-GEMM conv via V_WMMA_F32_16X16X4_F32 (exact fp32 path).
// Block = 256 threads = 8 wave32. wave&1 -> M-tile (targets 0-15 / 16-31),
// wave>>1 -> group of 4 consecutive N-tiles (64 time steps). Each wave keeps
// 4 accumulators and reuses one A fragment across 4 WMMAs per K-step, cutting
// weight (L2) traffic 4x vs one-tile-per-wave.
// y (+bias) stored at out[o*OUTW + t + 1]; scan phase overwrites in place.
// ---------------------------------------------------------------------------
__global__ __launch_bounds__(256) void conv_wmma_kernel(
    const float* __restrict__ x, const float* __restrict__ w,
    const float* __restrict__ bias, float* __restrict__ out)
{
    __shared__ float xs[NSRC * XSTRIDE];   // 41.5 KB
    const int tid  = threadIdx.x;
    const int lane = tid & 31;
    const int wave = tid >> 5;
    const int t0   = blockIdx.x * TILE_T;

    // Stage x tile: rows i = 0..31, cols j = 0..318 -> x[i][t0+j] (0-pad tail)
    for (int e = tid; e < NSRC * XCOLS; e += 256) {
        const int i  = e / XCOLS;
        const int j  = e - i * XCOLS;
        const int gt = t0 + j;
        xs[i * XSTRIDE + j] = (gt < T_LEN) ? x[i * T_LEN + gt] : 0.0f;
    }
    __syncthreads();

    const int mBase = (wave & 1) * 16;
    const int nBase = (wave >> 1) * 64;     // 4 N-tiles of 16
    const int ml    = lane & 15;            // M for A frag, N for B frag
    const int koff  = (lane >= 16) ? 2 : 0; // high half-lanes hold K+2,K+3

    // A: w[(mBase+ml)*2048 + k0 + koff + {0,1}] -> contiguous, 8B aligned
    const float* aPtr  = w + (size_t)(mBase + ml) * KDIM + koff;
    // B: xs[i*XSTRIDE + (k0&63) + koff + nBase + sub*16 + ml + {0,1}]
    const float* bBase = xs + koff + nBase + ml;

    v8f acc0 = {}, acc1 = {}, acc2 = {}, acc3 = {};

    #pragma unroll 4
    for (int k0 = 0; k0 < KDIM; k0 += 4) {
        const int i  = k0 >> 6;
        const int kk = k0 & 63;

        const float2 av = *reinterpret_cast<const float2*>(aPtr + k0);
        v2f a; a.x = av.x; a.y = av.y;

        const float* bp = bBase + i * XSTRIDE + kk;
        v2f b0; b0.x = bp[0];  b0.y = bp[1];
        v2f b1; b1.x = bp[16]; b1.y = bp[17];
        v2f b2; b2.x = bp[32]; b2.y = bp[33];
        v2f b3; b3.x = bp[48]; b3.y = bp[49];

        acc0 = __builtin_amdgcn_wmma_f32_16x16x4_f32(false, a, false, b0, (short)0, acc0, false, false);
        acc1 = __builtin_amdgcn_wmma_f32_16x16x4_f32(false, a, false, b1, (short)0, acc1, false, false);
        acc2 = __builtin_amdgcn_wmma_f32_16x16x4_f32(false, a, false, b2, (short)0, acc2, false, false);
        acc3 = __builtin_amdgcn_wmma_f32_16x16x4_f32(false, a, false, b3, (short)0, acc3, false, false);
    }

    // C/D layout: VGPR r -> M = r (lanes 0-15) or M = 8+r (lanes 16-31); N = lane&15
    v8f accs[4] = {acc0, acc1, acc2, acc3};
    #pragma unroll
    for (int sub = 0; sub < 4; ++sub) {
        const int tcol = t0 + nBase + sub * 16 + ml;
        if (tcol < L_OUT) {
            #pragma unroll
            for (int r = 0; r < 8; ++r) {
                const int o = mBase + ((lane < 16) ? r : (8 + r));
                out[(size_t)o * OUTW + tcol + 1] = accs[sub][r] + bias[o];
            }
        }
    }
}

// ---------------------------------------------------------------------------
// Scan parallelization: pos in {-1,0,+1} -> 3-state transition maps per chunk.
// ---------------------------------------------------------------------------
__device__ __forceinline__ float sgnf(float z) {
    return (z > 0.f) ? 1.f : ((z < 0.f) ? -1.f : 0.f);
}

// Phase 1: per (target, chunk) simulate all 3 entry states in one pass over y.
__global__ void scan_maps_kernel(const float* __restrict__ y,   // = d_out
                                 const float* __restrict__ pw,
                                 int8_t* __restrict__ maps)
{
    const int gid = blockIdx.x * blockDim.x + threadIdx.x;  // NTGT*CHUNKS
    const int o = gid >> 11;               // CHUNKS = 2048
    const int c = gid & (CHUNKS - 1);
    const float wv = pw[o];
    const float* yrow = y + (size_t)o * OUTW + 1;
    float pm = -1.f, pz = 0.f, pp = 1.f;
    const int tBeg = c * CLEN;
    const int tEnd = (tBeg + CLEN < L_OUT) ? (tBeg + CLEN) : L_OUT;
    for (int t = tBeg; t < tEnd; ++t) {
        const float yv = yrow[t];
        pm = sgnf(fmaf(pm, wv, yv));
        pz = sgnf(fmaf(pz, wv, yv));
        pp = sgnf(fmaf(pp, wv, yv));
    }
    int8_t* mp = maps + (size_t)(o * CHUNKS + c) * 3;
    mp[0] = (int8_t)pm; mp[1] = (int8_t)pz; mp[2] = (int8_t)pp;
}

// Phase 2: sequential composition over chunk boundaries (32 lanes, 2048 steps).
__global__ void scan_compose_kernel(const int8_t* __restrict__ maps,
                                    int8_t* __restrict__ entry)
{
    const int o = threadIdx.x;             // 32 threads
    int8_t pos = -1;
    for (int c = 0; c < CHUNKS; ++c) {
        entry[o * CHUNKS + c] = pos;
        pos = maps[(size_t)(o * CHUNKS + c) * 3 + (pos + 1)];
    }
}

// Phase 3: re-simulate each chunk from its known entry state; overwrite y
// in place with tanh(z). Same thread reads then writes each element.
__global__ void scan_emit_kernel(float* __restrict__ out,
                                 const float* __restrict__ pw,
                                 const int8_t* __restrict__ entry)
{
    const int gid = blockIdx.x * blockDim.x + threadIdx.x;
    const int o = gid >> 11;
    const int c = gid & (CHUNKS - 1);
    const float wv = pw[o];
    float pos = (float)entry[o * CHUNKS + c];
    if (c == 0) out[(size_t)o * OUTW] = -1.f;   // leading position
    float* yrow = out + (size_t)o * OUTW + 1;
    const int tBeg = c * CLEN;
    const int tEnd = (tBeg + CLEN < L_OUT) ? (tBeg + CLEN) : L_OUT;
    for (int t = tBeg; t < tEnd; ++t) {
        const float z = fmaf(pos, wv, yrow[t]);
        yrow[t] = tanhf(z);
        pos = sgnf(z);
    }
}

// ---------------------------------------------------------------------------
extern "C" void kernel_launch(void* const* d_in, const int* in_sizes, int n_in,
                              void* d_out, int out_size, void* d_ws, size_t ws_size,
                              hipStream_t stream)
{
    const float* x  = (const float*)d_in[0];   // (1,32,T)
    const float* cw = (const float*)d_in[1];   // (32,32,64)
    const float* cb = (const float*)d_in[2];   // (32,)
    const float* pw = (const float*)d_in[3];   // (32,1)
    float* out = (float*)d_out;                // (1,32,OUTW)

    int8_t* maps  = (int8_t*)d_ws;                              // 32*2048*3 B
    int8_t* entry = maps + (size_t)NTGT * CHUNKS * 3;           // 32*2048   B

    const int convBlocks = (L_OUT + TILE_T - 1) / TILE_T;       // 2048
    conv_wmma_kernel<<<convBlocks, 256, 0, stream>>>(x, cw, cb, out);

    const int scanBlocks = (NTGT * CHUNKS) / 256;               // 256
    scan_maps_kernel<<<scanBlocks, 256, 0, stream>>>(out, pw, maps);
    scan_compose_kernel<<<1, 32, 0, stream>>>(maps, entry);
    scan_emit_kernel<<<scanBlocks, 256, 0, stream>>>(out, pw, entry);
}